// Algebraic_5111011082842
// MI455X (gfx1250) — compile-verified
//
#include <hip/hip_runtime.h>
#include <stdint.h>
#include <stddef.h>

// Problem constants (match reference: ADD=3, N_COLS=16)
#define NCOL   16
#define NOUT   696            // 16 + C(16,2)=120 + C(16,3)=560
#define ROWS   16             // rows staged per block
#define RSTR   20             // padded LDS row stride in floats (80B -> 16B-aligned row bases)
#define QUADS  (NOUT / 4)     // 174 float4 stores per row (696 % 4 == 0)
#define BLOCK  256

typedef float    f32x4 __attribute__((ext_vector_type(4)));
typedef uint32_t u32x4 __attribute__((ext_vector_type(4)));

// ---- subset-index table, packed as 3 byte-offsets (i*4 | j*4<<8 | k*4<<16).
// Unused slots point at byte offset 64 == xs[row*RSTR + 16], which holds 1.0f.
// Order matches the reference exactly: identity cols, then lexicographic pairs,
// then lexicographic triples (itertools.combinations order).
struct Tab { uint32_t v[NOUT]; };

static constexpr Tab make_tab() {
  Tab t{};
  int p = 0;
  const uint32_t S = 16u * 4u;  // sentinel byte offset -> 1.0f
  for (uint32_t i = 0; i < NCOL; ++i)
    t.v[p++] = (i * 4u) | (S << 8) | (S << 16);
  for (uint32_t i = 0; i < NCOL; ++i)
    for (uint32_t j = i + 1; j < NCOL; ++j)
      t.v[p++] = (i * 4u) | ((j * 4u) << 8) | (S << 16);
  for (uint32_t i = 0; i < NCOL; ++i)
    for (uint32_t j = i + 1; j < NCOL; ++j)
      for (uint32_t k = j + 1; k < NCOL; ++k)
        t.v[p++] = (i * 4u) | ((j * 4u) << 8) | ((k * 4u) << 16);
  return t;
}

__constant__ __align__(16) Tab c_tab = make_tab();

// CDNA5 async global->LDS staging path (guarded; falls back to sync copy)
#if defined(__gfx1250__) && defined(__has_builtin)
#  if __has_builtin(__builtin_amdgcn_global_load_async_to_lds_b128)
#    define USE_ASYNC_LDS 1
#  endif
#endif
#ifndef USE_ASYNC_LDS
#  define USE_ASYNC_LDS 0
#endif

#if USE_ASYNC_LDS
// Matches the builtin's expected parameter types, as reported by clang:
//   arg0: __attribute__((vector_size(16))) int  __device__ *   (global / AS1)
//   arg1: LDS destination pointer                              (AS3)
typedef int v4i __attribute__((vector_size(16)));
typedef v4i __attribute__((address_space(1))) g_v4i;
typedef v4i __attribute__((address_space(3))) l_v4i;
#endif

__global__ __launch_bounds__(BLOCK) void poly_expand_kernel(
    const float* __restrict__ x, float* __restrict__ out, int nrows)
{
  __shared__ __align__(16) float    xs[ROWS * RSTR];
  __shared__ __align__(16) uint32_t ts[NOUT];

  const int tid = threadIdx.x;
  const int r0  = blockIdx.x * ROWS;

  // Copy packed subset table into LDS as 174 uint4s: one b128 load +
  // one ds_store_b128 per thread, no loop, no serialization.
  if (tid < QUADS)
    ((u32x4*)ts)[tid] = ((const u32x4*)c_tab.v)[tid];

  // Stage 16 rows x 16 floats of x into padded LDS rows: 64 lanes, one
  // 16-byte chunk each via async global->LDS B128 DMA. Lane l: row = l/4,
  // chunk = l%4 (chunks never cross a row; dest is 16B-aligned, RSTR=20).
  if (tid < ROWS * 4) {
    const int row  = tid >> 2;
    const int part = tid & 3;
    if (r0 + row < nrows) {
      const float* gsrc = x + (size_t)(r0 + row) * NCOL + part * 4;
      float*       ldst = &xs[row * RSTR + part * 4];
#if USE_ASYNC_LDS
      __builtin_amdgcn_global_load_async_to_lds_b128(
          (g_v4i*)gsrc, (l_v4i*)ldst, 0, 0);
#else
      *(f32x4*)ldst = *(const f32x4*)gsrc;
#endif
    }
  }
  if (tid < ROWS) xs[tid * RSTR + NCOL] = 1.0f;  // sentinel multiplier

  // Prefetch the next block's input tile into cache (global_prefetch_b8).
  if (tid == 0 && (r0 + ROWS) < nrows)
    __builtin_prefetch(x + (size_t)(r0 + ROWS) * NCOL, 0, 0);

#if USE_ASYNC_LDS
#  if __has_builtin(__builtin_amdgcn_s_wait_asynccnt)
  __builtin_amdgcn_s_wait_asynccnt(0);
#  else
  asm volatile("s_wait_asynccnt 0" ::: "memory");
#  endif
#endif
  __syncthreads();

  // Each thread emits one aligned float4 of output per iteration.
  // 16 rows * 174 quads = 2784 quads per block; stores are fully coalesced
  // (consecutive lanes -> consecutive 16B chunks) and non-temporal.
  for (int q = tid; q < ROWS * QUADS; q += BLOCK) {
    const int row  = q / QUADS;            // magic-number division
    const int quad = q - row * QUADS;
    const int grow = r0 + row;
    if (grow >= nrows) break;              // rows are nondecreasing in q

    const char* xb = (const char*)&xs[row * RSTR];
    const u32x4 t  = *(const u32x4*)&ts[quad * 4];

    f32x4 o;
    o.x = *(const float*)(xb + (t.x & 0xFFu)) *
          *(const float*)(xb + ((t.x >> 8) & 0xFFu)) *
          *(const float*)(xb + (t.x >> 16));
    o.y = *(const float*)(xb + (t.y & 0xFFu)) *
          *(const float*)(xb + ((t.y >> 8) & 0xFFu)) *
          *(const float*)(xb + (t.y >> 16));
    o.z = *(const float*)(xb + (t.z & 0xFFu)) *
          *(const float*)(xb + ((t.z >> 8) & 0xFFu)) *
          *(const float*)(xb + (t.z >> 16));
    o.w = *(const float*)(xb + (t.w & 0xFFu)) *
          *(const float*)(xb + ((t.w >> 8) & 0xFFu)) *
          *(const float*)(xb + (t.w >> 16));

    f32x4* dst = (f32x4*)(out + (size_t)grow * NOUT + quad * 4);
    __builtin_nontemporal_store(o, dst);   // streaming B128 store, NT hint
  }
}

extern "C" void kernel_launch(void* const* d_in, const int* in_sizes, int n_in,
                              void* d_out, int out_size, void* d_ws, size_t ws_size,
                              hipStream_t stream) {
  (void)n_in; (void)out_size; (void)d_ws; (void)ws_size;
  const float* x   = (const float*)d_in[0];
  float*       out = (float*)d_out;
  const int nrows  = in_sizes[0] / NCOL;           // 262144
  const int blocks = (nrows + ROWS - 1) / ROWS;    // 16384
  hipLaunchKernelGGL(poly_expand_kernel, dim3(blocks), dim3(BLOCK), 0, stream,
                     x, out, nrows);
}